// GpuCvxpyLayer_17755394802320
// MI455X (gfx1250) — compile-verified
//
#include <hip/hip_runtime.h>
#include <hip/hip_bf16.h>
#include <math.h>
#include <stdint.h>

typedef float v2f __attribute__((ext_vector_type(2)));
typedef float v8f __attribute__((ext_vector_type(8)));

#define BATCH 512
#define NDIM  200
#define MCON  100
#define KDIM  150
#define KP1   151
#define KPAD  152
#define NNZP  4000
#define NNZA  3000
#define RIDGE 5.0f

// gfx1250 async global->LDS copy (ASYNCcnt-tracked). VDST operand is the LDS
// byte address = low 32 bits of the generic shared pointer.
__device__ __forceinline__ void async_load_f32(float* lds_dst, const float* gsrc) {
  uint32_t laddr = (uint32_t)(uintptr_t)lds_dst;
  asm volatile("global_load_async_to_lds_b32 %0, %1, off"
               :: "v"(laddr), "v"(gsrc) : "memory");
}
__device__ __forceinline__ void wait_async_zero() {
  asm volatile("s_wait_asynccnt 0x0" ::: "memory");
}

// ---------------------------------------------------------------------------
// Kernel 1: eval[b,e] = sum_k pstack(b,k) * map[e*151+k]  via WMMA f32 16x16x4
// ---------------------------------------------------------------------------
__global__ __launch_bounds__(128) void eval_gemm_wmma(
    const float* __restrict__ params,
    const float* __restrict__ map,
    float* __restrict__ out, int E)
{
  __shared__ float sPS [64 * KPAD];
  __shared__ float sMap[16 * KPAD];

  const int tid   = threadIdx.x;
  const int e0    = blockIdx.x * 16;
  const int bbase = blockIdx.y * 64;

  for (int r = tid; r < 64; r += 128) {
    sPS[r * KPAD + KDIM]     = 1.0f;   // ones column
    sPS[r * KPAD + KDIM + 1] = 0.0f;   // zero pad
  }
  if (tid < 16) sMap[tid * KPAD + KP1] = 0.0f;

  for (int idx = tid; idx < 64 * KDIM; idx += 128) {
    int r = idx / KDIM, k = idx - r * KDIM;
    async_load_f32(&sPS[r * KPAD + k], &params[(size_t)(bbase + r) * KDIM + k]);
  }
  for (int idx = tid; idx < 16 * KP1; idx += 128) {
    int e = idx / KP1, k = idx - e * KP1;
    int ecol = e0 + e;
    if (ecol < E) async_load_f32(&sMap[e * KPAD + k], &map[(size_t)ecol * KP1 + k]);
    else          sMap[e * KPAD + k] = 0.0f;
  }
  wait_async_zero();
  __syncthreads();

  const int lane  = tid & 31;
  const int wid   = tid >> 5;
  const int khalf = lane >> 4;
  const int ml    = lane & 15;

  const float* pA = &sPS[(wid * 16 + ml) * KPAD];
  const float* pB = &sMap[ml * KPAD];

  v8f c = {};
  for (int k0 = 0; k0 < KPAD; k0 += 4) {
    const int ka = k0 + 2 * khalf;
    v2f a  = *reinterpret_cast<const v2f*>(pA + ka);
    v2f bb = *reinterpret_cast<const v2f*>(pB + ka);
    c = __builtin_amdgcn_wmma_f32_16x16x4_f32(false, a, false, bb,
                                              (short)0, c, false, false);
  }
  const int b0 = bbase + wid * 16;
  const int e  = e0 + ml;
#pragma unroll
  for (int i = 0; i < 8; ++i) {
    int m = i + 8 * khalf;
    if (e < E) out[(size_t)(b0 + m) * E + e] = c[i];
  }
}

// ---------------------------------------------------------------------------
// Kernel 2: per-batch KKT solve via Schur complement, entirely in LDS, with
// blocked (NB=16) Cholesky / trsm whose trailing updates run on WMMA.
// ---------------------------------------------------------------------------
#define PST   210        // even (aligned b64), 210%64=18 -> conflict-free cols
#define PROWS 208        // 13 tiles of 16 (pad rows/cols 200..207)
#define WST   210
#define WROWS 112        // 7 tiles of 16 (pad rows 100..111)
#define SST   114        // even, 114%64=50 -> conflict-free
#define SROWS 112
#define OFF_P   0
#define OFF_W   (OFF_P + PROWS * PST)        // 43680
#define OFF_S   (OFF_W + WROWS * WST)        // 67200
#define OFF_Q   (OFF_S + SROWS * SST)        // 79968
#define OFF_B   (OFF_Q + NDIM)               // 80168
#define OFF_Z   (OFF_B + 112)                // 80280
#define OFF_H   (OFF_Z + NDIM)               // 80480
#define OFF_NU  (OFF_H + NDIM)               // 80680
#define SMEM_FLOATS (OFF_NU + 112)           // 80792 floats = 323168 B < 320KB LDS

// Blocked Cholesky (lower), n = ntile*16, pad diagonals must be preset to 1.
// Panel factored scalar; trailing tiles updated with 4 chained WMMAs.
__device__ void chol_blocked(float* M, int stride, int ntile, int tid, int nt)
{
  const int n     = ntile * 16;
  const int lane  = tid & 31, wid = tid >> 5;
  const int khalf = lane >> 4, ml = lane & 15;
  const int nwave = nt >> 5;

  for (int p = 0; p < ntile; ++p) {
    const int jb = p * 16;
    // ---- panel factorization: columns jb..jb+15, rows j..n-1 ----
    for (int jc = 0; jc < 16; ++jc) {
      const int j = jb + jc;
      if (tid == 0) M[j * stride + j] = sqrtf(M[j * stride + j]);
      __syncthreads();
      const float d = M[j * stride + j];
      for (int i = j + 1 + tid; i < n; i += nt) M[i * stride + j] /= d;
      __syncthreads();
      const int ncc  = jb + 15 - j;
      const int nrow = n - (j + 1);
      for (int idx = tid; idx < ncc * nrow; idx += nt) {
        int cc = j + 1 + idx % ncc;
        int i  = j + 1 + idx / ncc;
        M[i * stride + cc] -= M[i * stride + j] * M[cc * stride + j];
      }
      __syncthreads();
    }
    // ---- trailing update (lower tiles only): C -= Lpanel * Lpanel^T ----
    const int T = ntile - 1 - p;
    const int njobs = T * (T + 1) / 2;
    for (int job = wid; job < njobs; job += nwave) {
      int a = 0;
      while ((a + 1) * (a + 2) / 2 <= job) ++a;       // uniform per wave
      const int ti = p + 1 + a;
      const int tj = p + 1 + (job - a * (a + 1) / 2);
      const int rI = ti * 16 + ml;
      const int rJ = tj * 16 + ml;
      v8f c;
#pragma unroll
      for (int i2 = 0; i2 < 8; ++i2)
        c[i2] = M[(ti * 16 + i2 + 8 * khalf) * stride + tj * 16 + ml];
#pragma unroll
      for (int ks = 0; ks < 4; ++ks) {
        const int ka = jb + ks * 4 + 2 * khalf;       // even -> aligned b64
        v2f a2 = *reinterpret_cast<const v2f*>(&M[rI * stride + ka]);
        v2f b2 = *reinterpret_cast<const v2f*>(&M[rJ * stride + ka]);
        a2 = -a2;                                     // f32 WMMA has no A-neg
        c = __builtin_amdgcn_wmma_f32_16x16x4_f32(false, a2, false, b2,
                                                  (short)0, c, false, false);
      }
#pragma unroll
      for (int i2 = 0; i2 < 8; ++i2)
        M[(ti * 16 + i2 + 8 * khalf) * stride + tj * 16 + ml] = c[i2];
    }
    __syncthreads();
  }
}

// Blocked trsm: Wm (nrw x n) <- Wm * L^-T, L lower (n x n). WMMA trailing.
__device__ void trsm_blocked(float* Wm, int wstride, int wtiles,
                             const float* L, int lstride, int ltiles,
                             int tid, int nt)
{
  const int nrw   = wtiles * 16;
  const int lane  = tid & 31, wid = tid >> 5;
  const int khalf = lane >> 4, ml = lane & 15;
  const int nwave = nt >> 5;

  for (int p = 0; p < ltiles; ++p) {
    const int jb = p * 16;
    // ---- solve against diagonal block ----
    for (int jc = 0; jc < 16; ++jc) {
      const int j = jb + jc;
      const float d = L[j * lstride + j];
      for (int r = tid; r < nrw; r += nt) Wm[r * wstride + j] /= d;
      __syncthreads();
      const int ncc = jb + 15 - j;
      for (int idx = tid; idx < ncc * nrw; idx += nt) {
        int cc = j + 1 + idx % ncc;
        int r  = idx / ncc;
        Wm[r * wstride + cc] -= L[cc * lstride + j] * Wm[r * wstride + j];
      }
      __syncthreads();
    }
    // ---- trailing: W[:, jb+16..] -= Wpanel * Lpanel^T ----
    const int T = ltiles - 1 - p;
    const int njobs = wtiles * T;
    for (int job = wid; job < njobs; job += nwave) {
      const int ti = job % wtiles;
      const int tj = p + 1 + job / wtiles;
      const int rI = ti * 16 + ml;
      const int rJ = tj * 16 + ml;
      v8f c;
#pragma unroll
      for (int i2 = 0; i2 < 8; ++i2)
        c[i2] = Wm[(ti * 16 + i2 + 8 * khalf) * wstride + tj * 16 + ml];
#pragma unroll
      for (int ks = 0; ks < 4; ++ks) {
        const int ka = jb + ks * 4 + 2 * khalf;
        v2f a2 = *reinterpret_cast<const v2f*>(&Wm[rI * wstride + ka]);
        v2f b2 = *reinterpret_cast<const v2f*>(&L[rJ * lstride + ka]);
        a2 = -a2;
        c = __builtin_amdgcn_wmma_f32_16x16x4_f32(false, a2, false, b2,
                                                  (short)0, c, false, false);
      }
#pragma unroll
      for (int i2 = 0; i2 < 8; ++i2)
        Wm[(ti * 16 + i2 + 8 * khalf) * wstride + tj * 16 + ml] = c[i2];
    }
    __syncthreads();
  }
}

__global__ __launch_bounds__(256) void kkt_solve(
    const float* __restrict__ Peval, const float* __restrict__ qeval,
    const float* __restrict__ Aeval,
    const int* __restrict__ Prow, const int* __restrict__ Pcol,
    const int* __restrict__ Arow, const int* __restrict__ Acol,
    float* __restrict__ out)
{
  extern __shared__ float sm[];
  float* P  = sm + OFF_P;
  float* W  = sm + OFF_W;   // holds A, then W = A L^-T in place
  float* S  = sm + OFF_S;
  float* q  = sm + OFF_Q;
  float* bv = sm + OFF_B;
  float* z  = sm + OFF_Z;
  float* h  = sm + OFF_H;
  float* nu = sm + OFF_NU;

  const int tid = threadIdx.x;
  const int nt  = blockDim.x;   // 256 = 8 wave32s
  const int b   = blockIdx.x;

  __builtin_prefetch(Prow, 0, 0);
  __builtin_prefetch(Pcol, 0, 0);

  // ---- zero LDS (pads stay zero forever) ----
  for (int i = tid; i < SMEM_FLOATS; i += nt) sm[i] = 0.0f;
  __syncthreads();

  // ---- load q, scatter P (symmetrized) and A / b into LDS ----
  for (int i = tid; i < NDIM; i += nt) q[i] = qeval[(size_t)b * NDIM + i];
  for (int i = tid; i < NNZP; i += nt) {
    float v = 0.5f * Peval[(size_t)b * NNZP + i];
    int r = Prow[i], cc = Pcol[i];
    atomicAdd(&P[r * PST + cc], v);
    atomicAdd(&P[cc * PST + r], v);
  }
  for (int i = tid; i < NNZA; i += nt) {
    float v = Aeval[(size_t)b * NNZA + i];
    int r = Arow[i], cc = Acol[i];
    if (cc < NDIM) atomicAdd(&W[r * WST + cc], v);   // A block
    else           atomicAdd(&bv[r], -v);            // b = -Ab[:,:,n]
  }
  __syncthreads();
  // ridge on real diagonal, 1.0 on pad diagonal (benign factorization)
  for (int j = tid; j < PROWS; j += nt) {
    if (j < NDIM) P[j * PST + j] += RIDGE;
    else          P[j * PST + j]  = 1.0f;
  }
  __syncthreads();

  // ---- P = L L^T (blocked, WMMA trailing updates) ----
  chol_blocked(P, PST, PROWS / 16, tid, nt);

  // ---- W = A L^-T (blocked trsm, WMMA trailing updates) ----
  trsm_blocked(W, WST, WROWS / 16, P, PST, PROWS / 16, tid, nt);

  // ---- S = W W^T (lower tiles), K = 208 incl. zero pads ----
  {
    const int lane  = tid & 31, wid = tid >> 5;
    const int khalf = lane >> 4, ml = lane & 15;
    for (int job = wid; job < 28; job += nt >> 5) {   // 7x7 lower tiles
      int a = 0;
      while ((a + 1) * (a + 2) / 2 <= job) ++a;
      const int ti = a;
      const int tj = job - a * (a + 1) / 2;
      const float* pA = &W[(ti * 16 + ml) * WST];
      const float* pB = &W[(tj * 16 + ml) * WST];
      v8f c = {};
      for (int k0 = 0; k0 < PROWS; k0 += 4) {
        const int ka = k0 + 2 * khalf;
        v2f a2 = *reinterpret_cast<const v2f*>(pA + ka);
        v2f b2 = *reinterpret_cast<const v2f*>(pB + ka);
        c = __builtin_amdgcn_wmma_f32_16x16x4_f32(false, a2, false, b2,
                                                  (short)0, c, false, false);
      }
#pragma unroll
      for (int i2 = 0; i2 < 8; ++i2)
        S[(ti * 16 + i2 + 8 * khalf) * SST + tj * 16 + ml] = c[i2];
    }
  }
  __syncthreads();
  for (int j = MCON + tid; j < SROWS; j += nt) S[j * SST + j] = 1.0f;
  __syncthreads();

  // ---- S = Ls Ls^T (blocked) ----
  chol_blocked(S, SST, SROWS / 16, tid, nt);

  // ---- z = L^-1 q ----
  for (int i = tid; i < NDIM; i += nt) z[i] = q[i];
  __syncthreads();
  for (int j = 0; j < NDIM; ++j) {
    if (tid == 0) z[j] /= P[j * PST + j];
    __syncthreads();
    float zj = z[j];
    for (int i = j + 1 + tid; i < NDIM; i += nt) z[i] -= P[i * PST + j] * zj;
    __syncthreads();
  }

  // ---- rhs_nu = -(b + W z) ----
  for (int r = tid; r < MCON; r += nt) {
    float acc = 0.0f;
    for (int cc = 0; cc < NDIM; ++cc) acc += W[r * WST + cc] * z[cc];
    nu[r] = -(bv[r] + acc);
  }
  __syncthreads();

  // ---- solve S nu = rhs_nu : forward (Ls) then backward (Ls^T) ----
  for (int j = 0; j < MCON; ++j) {
    if (tid == 0) nu[j] /= S[j * SST + j];
    __syncthreads();
    float vj = nu[j];
    for (int i = j + 1 + tid; i < MCON; i += nt) nu[i] -= S[i * SST + j] * vj;
    __syncthreads();
  }
  for (int j = MCON - 1; j >= 0; --j) {
    if (tid == 0) nu[j] /= S[j * SST + j];
    __syncthreads();
    float vj = nu[j];
    for (int i = tid; i < j; i += nt) nu[i] -= S[j * SST + i] * vj;
    __syncthreads();
  }

  // ---- t = -(z + W^T nu) ; x = L^-T t ----
  for (int cc = tid; cc < NDIM; cc += nt) {
    float acc = 0.0f;
    for (int r = 0; r < MCON; ++r) acc += W[r * WST + cc] * nu[r];
    h[cc] = -(z[cc] + acc);
  }
  __syncthreads();
  for (int j = NDIM - 1; j >= 0; --j) {
    if (tid == 0) h[j] /= P[j * PST + j];
    __syncthreads();
    float vj = h[j];
    for (int i = tid; i < j; i += nt) h[i] -= P[j * PST + i] * vj;
    __syncthreads();
  }

  // ---- outputs: x (B x 200) then nu (B x 100), concatenated flat ----
  for (int i = tid; i < NDIM; i += nt) out[(size_t)b * NDIM + i] = h[i];
  for (int r = tid; r < MCON; r += nt)
    out[(size_t)BATCH * NDIM + (size_t)b * MCON + r] = nu[r];
}

// ---------------------------------------------------------------------------
extern "C" void kernel_launch(void* const* d_in, const int* in_sizes, int n_in,
                              void* d_out, int out_size, void* d_ws, size_t ws_size,
                              hipStream_t stream)
{
  const float* params = (const float*)d_in[0];
  const float* Pmap   = (const float*)d_in[1];
  const float* qmap   = (const float*)d_in[2];
  const float* Amap   = (const float*)d_in[3];
  const int*   Prow   = (const int*)d_in[4];
  const int*   Pcol   = (const int*)d_in[5];
  const int*   Arow   = (const int*)d_in[6];
  const int*   Acol   = (const int*)d_in[7];
  (void)in_sizes; (void)n_in; (void)out_size; (void)ws_size;

  float* Peval = (float*)d_ws;                         // 512*4000
  float* qeval = Peval + (size_t)BATCH * NNZP;         // 512*200
  float* Aeval = qeval + (size_t)BATCH * NDIM;         // 512*3000

  dim3 blk(128);
  eval_gemm_wmma<<<dim3((NNZP + 15) / 16, BATCH / 64), blk, 0, stream>>>(params, Pmap, Peval, NNZP);
  eval_gemm_wmma<<<dim3((NDIM + 15) / 16, BATCH / 64), blk, 0, stream>>>(params, qmap, qeval, NDIM);
  eval_gemm_wmma<<<dim3((NNZA + 15) / 16, BATCH / 64), blk, 0, stream>>>(params, Amap, Aeval, NNZA);

  size_t smem = (size_t)SMEM_FLOATS * sizeof(float);   // ~316 KB < 320 KB WGP LDS
  kkt_solve<<<dim3(BATCH), dim3(256), smem, stream>>>(Peval, qeval, Aeval,
                                                      Prow, Pcol, Arow, Acol,
                                                      (float*)d_out);
}